// GCN_LSTM_11510512353639
// MI455X (gfx1250) — compile-verified
//
#include <hip/hip_runtime.h>
#include <hip/hip_bf16.h>
#include <math.h>

typedef float v2f __attribute__((ext_vector_type(2)));
typedef float v8f __attribute__((ext_vector_type(8)));

#define N_NODES_F 64   // H_GCN feature width used by both propagation passes
#define H_LSTM 128
#define N_CLS 10
#define N_GRAPHS 64

// ---------------- fp32 WMMA GEMM: C[M x N] = A[M x K] * B[K x N] ----------------
// N fixed at 64 (4 N-tiles), 8 waves per block -> 2 M-tiles x 4 N-tiles per block.
template<int K, int N>
__global__ void gemm_wmma_f32_kernel(const float* __restrict__ A,
                                     const float* __restrict__ B,
                                     float* __restrict__ C, int M) {
    __shared__ float Bs[K * N];
    for (int i = threadIdx.x; i < K * N; i += blockDim.x) Bs[i] = B[i];
    __syncthreads();

    const int lane = threadIdx.x & 31;
    const int wave = threadIdx.x >> 5;
    const int nt   = wave & 3;          // N tile index (N/16 == 4)
    const int mt   = wave >> 2;         // 0..1
    const int m0   = (blockIdx.x * 2 + mt) * 16;
    if (m0 + 16 > M) return;            // wave-uniform guard (EXEC stays all-1s)
    const int n0   = nt * 16;
    const int half = lane >> 4;         // 0 | 1
    const int r    = lane & 15;

    const float* arow = A + (size_t)(m0 + r) * K;
    v8f c = {};
    #pragma unroll 4
    for (int k0 = 0; k0 < K; k0 += 4) {
        const int ka = k0 + 2 * half;
        v2f a, b;
        a.x = arow[ka];
        a.y = arow[ka + 1];
        b.x = Bs[(size_t)ka * N + n0 + r];
        b.y = Bs[(size_t)(ka + 1) * N + n0 + r];
        c = __builtin_amdgcn_wmma_f32_16x16x4_f32(
                /*neg_a=*/false, a, /*neg_b=*/false, b,
                /*c_mod=*/(short)0, c, /*reuse_a=*/false, /*reuse_b=*/false);
    }
    float* crow = C + (size_t)m0 * N + n0 + r;
    #pragma unroll
    for (int j = 0; j < 8; ++j)
        crow[(size_t)(j + 8 * half) * N] = c[j];
}

// ---------------- degree count (edges only; self-loop folded in later) ----------
__global__ void deg_kernel(const int* __restrict__ dst, float* __restrict__ deg, int E) {
    int e = blockIdx.x * blockDim.x + threadIdx.x;
    if (e < E) atomicAdd(&deg[dst[e]], 1.0f);
}

__global__ void dinv_kernel(const float* __restrict__ deg, float* __restrict__ dinv, int n) {
    int i = blockIdx.x * blockDim.x + threadIdx.x;
    if (i < n) dinv[i] = rsqrtf(deg[i] + 1.0f);   // +1 for the self-loop
}

// ---------------- edge scatter: agg[d] += h[s] * dinv[s]*dinv[d] ----------------
// 16 lanes per edge, float4 per lane (F = 64)
__global__ void scatter_edges_kernel(const float* __restrict__ h,
                                     const int* __restrict__ src,
                                     const int* __restrict__ dst,
                                     const float* __restrict__ dinv,
                                     float* __restrict__ agg, int E) {
    int gid = blockIdx.x * blockDim.x + threadIdx.x;
    int e = gid >> 4;
    if (e >= E) return;
    int q = gid & 15;
    int s = src[e], d = dst[e];
    float norm = dinv[s] * dinv[d];
    float4 v = reinterpret_cast<const float4*>(h)[(size_t)s * 16 + q];
    float* out = agg + (size_t)d * 64 + q * 4;
    atomicAdd(out + 0, v.x * norm);
    atomicAdd(out + 1, v.y * norm);
    atomicAdd(out + 2, v.z * norm);
    atomicAdd(out + 3, v.w * norm);
}

// -------- self-loop contribution + bias + relu, in place into agg ---------------
__global__ void selfloop_bias_relu_kernel(const float* __restrict__ h,
                                          const float* __restrict__ bias,
                                          const float* __restrict__ dinv,
                                          float* __restrict__ agg, int n) {
    int gid = blockIdx.x * blockDim.x + threadIdx.x;
    if (gid >= n * 64) return;
    int i = gid >> 6, f = gid & 63;
    float di = dinv[i];
    float v = agg[gid] + h[gid] * di * di + bias[f];
    agg[gid] = v > 0.0f ? v : 0.0f;
}

// ---------------- global mean pool (sum + count via atomics) --------------------
__global__ void pool_kernel(const float* __restrict__ h,
                            const int* __restrict__ batch,
                            float* __restrict__ sums, float* __restrict__ cnt, int n) {
    int gid = blockIdx.x * blockDim.x + threadIdx.x;
    if (gid >= n * 64) return;
    int i = gid >> 6, f = gid & 63;
    int b = batch[i];
    atomicAdd(&sums[(size_t)b * 64 + f], h[gid]);
    if (f == 0) atomicAdd(&cnt[b], 1.0f);
}

// ---------------- single-step LSTM (h0=c0=0): one block per graph ---------------
__global__ void lstm_kernel(const float* __restrict__ sums,
                            const float* __restrict__ cnt,
                            const float* __restrict__ w_ih,
                            const float* __restrict__ b_ih,
                            const float* __restrict__ b_hh,
                            float* __restrict__ hh) {
    __shared__ float gm[64];
    __shared__ float gates[4 * H_LSTM];
    const int g = blockIdx.x;
    const int j = threadIdx.x;            // 0..511
    if (j < 64) {
        float c = cnt[g];
        gm[j] = sums[(size_t)g * 64 + j] / fmaxf(c, 1.0f);
    }
    __syncthreads();
    const float* wrow = w_ih + (size_t)j * 64;
    float acc = b_ih[j] + b_hh[j];
    #pragma unroll 8
    for (int k = 0; k < 64; ++k) acc += gm[k] * wrow[k];
    gates[j] = acc;
    __syncthreads();
    if (j < H_LSTM) {
        float ig = gates[j];
        float gg = gates[2 * H_LSTM + j];
        float og = gates[3 * H_LSTM + j];
        float si = 1.0f / (1.0f + expf(-ig));
        float so = 1.0f / (1.0f + expf(-og));
        float c  = si * tanhf(gg);        // c0 = 0 -> forget gate contributes 0
        hh[(size_t)g * H_LSTM + j] = so * tanhf(c);
    }
}

// ---------------- final FC: out[g][c] = hh[g] . fc_w[c] + fc_b[c] ---------------
__global__ void fc_kernel(const float* __restrict__ hh,
                          const float* __restrict__ fc_w,
                          const float* __restrict__ fc_b,
                          float* __restrict__ out) {
    const int g = blockIdx.x;
    const int t = threadIdx.x;
    if (t >= N_CLS) return;
    const float* hr = hh + (size_t)g * H_LSTM;
    const float* wr = fc_w + (size_t)t * H_LSTM;
    float acc = fc_b[t];
    #pragma unroll 8
    for (int k = 0; k < H_LSTM; ++k) acc += hr[k] * wr[k];
    out[(size_t)g * N_CLS + t] = acc;
}

extern "C" void kernel_launch(void* const* d_in, const int* in_sizes, int n_in,
                              void* d_out, int out_size, void* d_ws, size_t ws_size,
                              hipStream_t stream) {
    const float* x      = (const float*)d_in[0];
    const int*   eidx   = (const int*)  d_in[1];
    const int*   batch  = (const int*)  d_in[2];
    const float* w1     = (const float*)d_in[3];
    const float* b1     = (const float*)d_in[4];
    const float* w2     = (const float*)d_in[5];
    const float* b2     = (const float*)d_in[6];
    const float* w_ih   = (const float*)d_in[7];
    // d_in[8] = w_hh (unused: h0 = 0)
    const float* b_ih   = (const float*)d_in[9];
    const float* b_hh   = (const float*)d_in[10];
    const float* fc_w   = (const float*)d_in[11];
    const float* fc_b   = (const float*)d_in[12];
    float* out = (float*)d_out;

    const int Nn = in_sizes[0] / 128;     // 50000 nodes
    const int E  = in_sizes[1] / 2;       // 1.6M edges
    const int* src = eidx;
    const int* dst = eidx + E;
    const size_t NF = (size_t)Nn * 64;

    // workspace layout (floats). Zero-init region is contiguous -> one memset.
    float* ws   = (float*)d_ws;
    float* h1   = ws;                                 // NF
    float* h2   = h1 + NF;                            // NF
    float* dinv = h2 + NF;                            // Nn
    float* hh   = dinv + Nn;                          // 64*128
    float* zreg = hh + (size_t)N_GRAPHS * H_LSTM;     // zero region start
    float* agg1 = zreg;                               // NF
    float* agg2 = agg1 + NF;                          // NF
    float* deg  = agg2 + NF;                          // Nn
    float* sums = deg + Nn;                           // 64*64
    float* cnt  = sums + (size_t)N_GRAPHS * 64;       // 64
    const size_t zbytes = (size_t)(cnt + N_GRAPHS - zreg) * sizeof(float);
    hipMemsetAsync(zreg, 0, zbytes, stream);

    // degrees and normalization
    deg_kernel<<<(E + 255) / 256, 256, 0, stream>>>(dst, deg, E);
    dinv_kernel<<<(Nn + 255) / 256, 256, 0, stream>>>(deg, dinv, Nn);

    // layer 1: h1 = x @ w1 ; propagate ; + self-loop + b1 ; relu (in agg1)
    {
        int mblocks = (Nn / 16 + 1) / 2;
        gemm_wmma_f32_kernel<128, 64><<<mblocks, 256, 0, stream>>>(x, w1, h1, Nn);
    }
    scatter_edges_kernel<<<((E * 16) + 255) / 256, 256, 0, stream>>>(h1, src, dst, dinv, agg1, E);
    selfloop_bias_relu_kernel<<<((Nn * 64) + 255) / 256, 256, 0, stream>>>(h1, b1, dinv, agg1, Nn);

    // layer 2: h2 = agg1 @ w2 ; propagate ; + self-loop + b2 ; relu (in agg2)
    {
        int mblocks = (Nn / 16 + 1) / 2;
        gemm_wmma_f32_kernel<64, 64><<<mblocks, 256, 0, stream>>>(agg1, w2, h2, Nn);
    }
    scatter_edges_kernel<<<((E * 16) + 255) / 256, 256, 0, stream>>>(h2, src, dst, dinv, agg2, E);
    selfloop_bias_relu_kernel<<<((Nn * 64) + 255) / 256, 256, 0, stream>>>(h2, b2, dinv, agg2, Nn);

    // mean pool -> LSTM -> FC
    pool_kernel<<<((Nn * 64) + 255) / 256, 256, 0, stream>>>(agg2, batch, sums, cnt, Nn);
    lstm_kernel<<<N_GRAPHS, 4 * H_LSTM, 0, stream>>>(sums, cnt, w_ih, b_ih, b_hh, hh);
    fc_kernel<<<N_GRAPHS, 32, 0, stream>>>(hh, fc_w, fc_b, out);
}